// GraphMemory_11897059410437
// MI455X (gfx1250) — compile-verified
//
#include <hip/hip_runtime.h>
#include <stdint.h>

#define HDIM 384
#define WARP 32

typedef float v2f __attribute__((ext_vector_type(2)));
typedef float v8f __attribute__((ext_vector_type(8)));

__device__ __forceinline__ float waveSum(float v) {
#pragma unroll
  for (int off = 16; off > 0; off >>= 1) v += __shfl_xor(v, off, WARP);
  return v;
}

// monotonic float <-> uint key for atomic float max via atomicMax(u32)
__device__ __forceinline__ unsigned fkey(float f) {
  unsigned u = __float_as_uint(f);
  return (u & 0x80000000u) ? ~u : (u | 0x80000000u);
}
__device__ __forceinline__ float unkey(unsigned k) {
  unsigned u = (k & 0x80000000u) ? (k & 0x7fffffffu) : ~k;
  return __uint_as_float(u);
}

// ---------------- prep: v_edge = W_edge @ att_edge ; qnorm ----------------
__global__ void prep_kernel(const float* __restrict__ W_edge,
                            const float* __restrict__ att_edge,
                            const float* __restrict__ query,
                            float* __restrict__ v_edge,
                            float* __restrict__ scal) {
  __shared__ float sh[512];
  int tid = threadIdx.x;
  float q = 0.f;
  if (tid < HDIM) {
    float s = 0.f;
    const float* row = W_edge + (size_t)tid * HDIM;
#pragma unroll 8
    for (int j = 0; j < HDIM; ++j) s += row[j] * att_edge[j];
    v_edge[tid] = s;
    q = query[tid] * query[tid];
  }
  sh[tid] = q;
  __syncthreads();
  for (int off = 256; off > 0; off >>= 1) {
    if (tid < off) sh[tid] += sh[tid + off];
    __syncthreads();
  }
  if (tid == 0) scal[0] = fmaxf(sqrtf(sh[0]), 1e-8f);
}

// ---------------- init: out = bias (broadcast), segkey=0, denom=0 ----------
__global__ void init_kernel(float* __restrict__ outb, unsigned* __restrict__ segkey,
                            float* __restrict__ denom, const float* __restrict__ bias,
                            int N) {
  int i = blockIdx.x * blockDim.x + threadIdx.x;
  size_t total = (size_t)N * HDIM;
  if ((size_t)i < total) outb[i] = bias[i % HDIM];
  if (i < N) { segkey[i] = 0u; denom[i] = 0.f; }
}

// ---------------- h = node_emb @ W via V_WMMA_F32_16X16X4_F32 --------------
// One wave computes one 16x16 tile of C. A: 16x4 f32 frag (2 VGPR),
// B: 4x16 f32 frag (2 VGPR), C/D: 8 VGPR f32.
__global__ void gemm_h_wmma(const float* __restrict__ A, const float* __restrict__ B,
                            float* __restrict__ C, int Nrows) {
  const int wave = (blockIdx.x * blockDim.x + threadIdx.x) >> 5;
  const int lane = threadIdx.x & 31;
  const int tilesN = HDIM >> 4;                 // 24
  const int tm = wave / tilesN;
  const int tn = wave - tm * tilesN;
  if (tm * 16 >= Nrows) return;
  const int m0 = tm << 4, n0 = tn << 4;
  const int hs = lane >> 4;                     // half-select 0/1 (K pairs 0,1 / 2,3)
  const int l16 = lane & 15;

  const float* arow = A + (size_t)(m0 + l16) * HDIM + 2 * hs;   // A[m, k+2*hs]
  const float* bcol = B + (size_t)(2 * hs) * HDIM + (n0 + l16); // B[k+2*hs, n]

  v8f acc = {};
#pragma unroll 4
  for (int k = 0; k < HDIM; k += 4) {
    v2f a, b;
    a.x = arow[k];
    a.y = arow[k + 1];
    b.x = bcol[(size_t)k * HDIM];
    b.y = bcol[(size_t)(k + 1) * HDIM];
    acc = __builtin_amdgcn_wmma_f32_16x16x4_f32(false, a, false, b,
                                                (short)0, acc, false, false);
  }
  // C tile: lane holds column n0+l16, rows m0 + r + 8*hs (r=0..7)
#pragma unroll
  for (int r = 0; r < 8; ++r)
    C[(size_t)(m0 + r + 8 * hs) * HDIM + (n0 + l16)] = acc[r];
}

// ---------------- a_src/a_dst: per-node dots with att vectors --------------
__global__ void asrc_adst_kernel(const float* __restrict__ h,
                                 const float* __restrict__ att_src,
                                 const float* __restrict__ att_dst,
                                 float* __restrict__ a_src, float* __restrict__ a_dst,
                                 int N) {
  int wave = (blockIdx.x * blockDim.x + threadIdx.x) >> 5;
  int lane = threadIdx.x & 31;
  if (wave >= N) return;
  const float* row = h + (size_t)wave * HDIM;
  float s1 = 0.f, s2 = 0.f;
#pragma unroll
  for (int t = 0; t < HDIM / WARP; ++t) {
    int j = lane + t * WARP;
    float x = row[j];
    s1 += x * att_src[j];
    s2 += x * att_dst[j];
  }
  s1 = waveSum(s1);
  s2 = waveSum(s2);
  if (lane == 0) { a_src[wave] = s1; a_dst[wave] = s2; }
}

// ---------------- a_edge[e] = edge_attr[e,:] . v_edge  (HBM roofline) ------
__global__ void aedge_kernel(const float* __restrict__ EA, const float* __restrict__ v,
                             float* __restrict__ a_edge, int E) {
  int wave = (blockIdx.x * blockDim.x + threadIdx.x) >> 5;
  int lane = threadIdx.x & 31;
  if (wave >= E) return;
  const float4* row = (const float4*)(EA + (size_t)wave * HDIM);
  const float4* vv = (const float4*)v;
  float s = 0.f;
#pragma unroll
  for (int t = 0; t < HDIM / (4 * WARP); ++t) {  // 3 x float4 per lane
    float4 x = row[lane + t * WARP];
    float4 y = vv[lane + t * WARP];
    s += x.x * y.x + x.y * y.y + x.z * y.z + x.w * y.w;
  }
  s = waveSum(s);
  if (lane == 0) a_edge[wave] = s;
}

// ---------------- mean of a_edge -> scal[1] (self-loop attention) ----------
__global__ void mean_aedge_kernel(const float* __restrict__ a_edge, float* __restrict__ scal,
                                  int E) {
  __shared__ float sh[256];
  int tid = threadIdx.x;
  float s = 0.f;
  for (int i = tid; i < E; i += 256) s += a_edge[i];
  sh[tid] = s;
  __syncthreads();
  for (int off = 128; off > 0; off >>= 1) {
    if (tid < off) sh[tid] += sh[tid + off];
    __syncthreads();
  }
  if (tid == 0) scal[1] = sh[0] / (float)E;
}

// ---------------- logits + segment max (atomic u32 key max) ----------------
__global__ void logits_segmax_kernel(const int* __restrict__ ei_src, const int* __restrict__ ei_dst,
                                     const float* __restrict__ a_src, const float* __restrict__ a_dst,
                                     const float* __restrict__ a_edge, const float* __restrict__ scal,
                                     float* __restrict__ logitw, unsigned* __restrict__ segkey,
                                     int E, int N) {
  int i = blockIdx.x * blockDim.x + threadIdx.x;
  int M = E + N;
  if (i >= M) return;
  int s, d; float ae;
  if (i < E) { s = ei_src[i]; d = ei_dst[i]; ae = a_edge[i]; }
  else       { s = d = i - E; ae = scal[1]; }
  float l = a_src[s] + a_dst[d] + ae;
  l = (l > 0.f) ? l : 0.2f * l;                 // leaky_relu(0.2)
  logitw[i] = l;
  atomicMax(&segkey[d], fkey(l));
}

// ---------------- w = exp(l - segmax[dst]); denom[dst] += w ---------------
__global__ void weights_kernel(const int* __restrict__ ei_dst,
                               const unsigned* __restrict__ segkey,
                               float* __restrict__ logitw, float* __restrict__ denom,
                               int E, int N) {
  int i = blockIdx.x * blockDim.x + threadIdx.x;
  int M = E + N;
  if (i >= M) return;
  int d = (i < E) ? ei_dst[i] : (i - E);
  float w = __expf(logitw[i] - unkey(segkey[d]));
  logitw[i] = w;
  atomicAdd(&denom[d], w);
}

// ---------------- out[dst] += alpha * h[src]  (wave per edge) --------------
__global__ void scatter_kernel(const int* __restrict__ ei_src, const int* __restrict__ ei_dst,
                               const float* __restrict__ h, const float* __restrict__ logitw,
                               const float* __restrict__ denom, float* __restrict__ outb,
                               int E, int N) {
  int wave = (blockIdx.x * blockDim.x + threadIdx.x) >> 5;
  int lane = threadIdx.x & 31;
  int M = E + N;
  if (wave >= M) return;
  int s, d;
  if (wave < E) { s = ei_src[wave]; d = ei_dst[wave]; }
  else          { s = d = wave - E; }
  float alpha = logitw[wave] / denom[d];
  const float* hs = h + (size_t)s * HDIM;
  float* od = outb + (size_t)d * HDIM;
#pragma unroll
  for (int t = 0; t < HDIM / WARP; ++t) {
    int j = lane + t * WARP;
    atomicAdd(&od[j], hs[j] * alpha);
  }
}

// ---------------- cosine similarity per node -------------------------------
__global__ void sim_kernel(const float* __restrict__ outb, const float* __restrict__ query,
                           const float* __restrict__ scal, float* __restrict__ sim, int N) {
  int wave = (blockIdx.x * blockDim.x + threadIdx.x) >> 5;
  int lane = threadIdx.x & 31;
  if (wave >= N) return;
  const float* row = outb + (size_t)wave * HDIM;
  float nn = 0.f, qd = 0.f;
#pragma unroll
  for (int t = 0; t < HDIM / WARP; ++t) {
    int j = lane + t * WARP;
    float x = row[j];
    nn += x * x;
    qd += x * query[j];
  }
  nn = waveSum(nn);
  qd = waveSum(qd);
  if (lane == 0) sim[wave] = qd / (fmaxf(sqrtf(nn), 1e-8f) * scal[0]);
}

// ---------------- argmax (first-max tie-break) + emit out[idx] -------------
__global__ void argmax_kernel(const float* __restrict__ sim, const float* __restrict__ outb,
                              float* __restrict__ d_out, int N) {
  __shared__ float sv[256];
  __shared__ int si[256];
  int tid = threadIdx.x;
  float best = -3.402823466e+38f;
  int bi = 0;
  for (int i = tid; i < N; i += 256) {
    float s = sim[i];
    if (s > best) { best = s; bi = i; }
  }
  sv[tid] = best; si[tid] = bi;
  __syncthreads();
  for (int off = 128; off > 0; off >>= 1) {
    if (tid < off) {
      if (sv[tid + off] > sv[tid] ||
          (sv[tid + off] == sv[tid] && si[tid + off] < si[tid])) {
        sv[tid] = sv[tid + off];
        si[tid] = si[tid + off];
      }
    }
    __syncthreads();
  }
  int idx = si[0];
  for (int j = tid; j < HDIM; j += 256) d_out[j] = outb[(size_t)idx * HDIM + j];
}

extern "C" void kernel_launch(void* const* d_in, const int* in_sizes, int n_in,
                              void* d_out, int out_size, void* d_ws, size_t ws_size,
                              hipStream_t stream) {
  (void)n_in; (void)out_size; (void)ws_size;
  const float* query    = (const float*)d_in[0];
  const float* node_emb = (const float*)d_in[1];
  const float* edge_attr= (const float*)d_in[2];
  const float* W        = (const float*)d_in[3];
  const float* att_src  = (const float*)d_in[4];
  const float* att_dst  = (const float*)d_in[5];
  const float* W_edge   = (const float*)d_in[6];
  const float* att_edge = (const float*)d_in[7];
  const float* bias     = (const float*)d_in[8];
  const int*   ei       = (const int*)d_in[9];

  const int Hh = in_sizes[0];              // 384
  const int N  = in_sizes[1] / Hh;         // 10000
  const int E  = in_sizes[2] / Hh;         // 320000
  const int M  = E + N;
  const int* ei_src = ei;
  const int* ei_dst = ei + E;

  // workspace layout (floats)
  float* ws      = (float*)d_ws;
  float* h       = ws;                          // N*H
  float* outb    = h + (size_t)N * Hh;          // N*H
  float* a_src   = outb + (size_t)N * Hh;       // N
  float* a_dst   = a_src + N;                   // N
  unsigned* segk = (unsigned*)(a_dst + N);      // N
  float* denom   = (float*)(segk + N);          // N
  float* sim     = denom + N;                   // N
  float* a_edge  = sim + N;                     // E
  float* logitw  = a_edge + E;                  // E+N
  float* v_edge  = logitw + M;                  // H
  float* scal    = v_edge + Hh;                 // [0]=qnorm, [1]=mean(a_edge)

  // 1. v_edge = W_edge @ att_edge ; qnorm
  prep_kernel<<<1, 512, 0, stream>>>(W_edge, att_edge, query, v_edge, scal);

  // 2. out = bias broadcast; segkey/denom = 0
  {
    size_t total = (size_t)N * Hh;
    init_kernel<<<(unsigned)((total + 255) / 256), 256, 0, stream>>>(outb, segk, denom, bias, N);
  }

  // 3. h = node_emb @ W  (WMMA f32 16x16x4; 625 x 24 tiles, 1 wave/tile)
  {
    int waves = (N / 16) * (Hh / 16);
    gemm_h_wmma<<<(waves * 32 + 255) / 256, 256, 0, stream>>>(node_emb, W, h, N);
  }

  // 4. a_src / a_dst per node
  asrc_adst_kernel<<<(N * 32 + 255) / 256, 256, 0, stream>>>(h, att_src, att_dst, a_src, a_dst, N);

  // 5. a_edge per edge (dominant: 491 MB stream)
  {
    long long waves = (long long)E;
    aedge_kernel<<<(unsigned)((waves * 32 + 255) / 256), 256, 0, stream>>>(edge_attr, v_edge, a_edge, E);
  }

  // 6. mean(a_edge) -> scal[1]
  mean_aedge_kernel<<<1, 256, 0, stream>>>(a_edge, scal, E);

  // 7. logits + segment max
  logits_segmax_kernel<<<(M + 255) / 256, 256, 0, stream>>>(ei_src, ei_dst, a_src, a_dst,
                                                            a_edge, scal, logitw, segk, E, N);

  // 8. exp weights + denom
  weights_kernel<<<(M + 255) / 256, 256, 0, stream>>>(ei_dst, segk, logitw, denom, E, N);

  // 9. scatter-add aggregation (wave per edge)
  {
    long long waves = (long long)M;
    scatter_kernel<<<(unsigned)((waves * 32 + 255) / 256), 256, 0, stream>>>(
        ei_src, ei_dst, h, logitw, denom, outb, E, N);
  }

  // 10. cosine similarity per node
  sim_kernel<<<(N * 32 + 255) / 256, 256, 0, stream>>>(outb, query, scal, sim, N);

  // 11. argmax + write result row
  argmax_kernel<<<1, 256, 0, stream>>>(sim, outb, (float*)d_out, N);
}